// cutmix_33457795236027
// MI455X (gfx1250) — compile-verified
//
#include <hip/hip_runtime.h>
#include <stdint.h>

// Problem constants (match the reference).
#define NB 128
#define NC 3
#define NH 224
#define NW 224
#define NK 1000
#define W4 (NW / 4)                       // 56 float4 per row
#define PLANE (NH * NW)                   // 50176 floats per (b,ch) plane
#define IMG_FLOATS (NB * NC * NH * NW)    // 19,267,584
#define LBL_FLOATS (NB * NK)              // 128,000

typedef float v4f __attribute__((ext_vector_type(4)));

// ---------------------------------------------------------------------------
// Images: one 64-thread (2-wave) block per (b, h) row, covering all 3
// channels. The box decision is block-uniform -> scalar branch.
//   Fast path (~75-80% of rows): DMA copy global -> LDS -> global with the
//   CDNA5 async-LDS engine; 3 loads queued on ASYNCcnt before one wait, then
//   3 stores queued back-to-back (drained by S_ENDPGM's implicit wait-idle).
//   Blend path: per-column 128-bit select; column mask computed once and
//   reused across channels; non-temporal stores (output is never re-read).
// ---------------------------------------------------------------------------
__global__ __launch_bounds__(64) void cutmix_images_kernel(
    const float* __restrict__ images,
    const int*   __restrict__ perm,
    const int*   __restrict__ box_size,
    const int*   __restrict__ xarr,
    const int*   __restrict__ yarr,
    const unsigned char* __restrict__ keep,
    float* __restrict__ out)
{
    __shared__ v4f stage[NC][64];

    const int row = blockIdx.x;            // 0 .. NB*NH-1
    const int b   = row / NH;              // uniform -> scalar ops
    const int h   = row - b * NH;

    const int  lane   = threadIdx.x;
    const bool active = lane < W4;

    // Per-sample metadata: b is uniform, compiler scalarizes these loads.
    const int bx = xarr[b];
    const int bs = box_size[b];
    const int by = yarr[b];
    const int kp = (int)keep[b];
    const int pb = perm[b];

    const size_t base = (size_t)b * (NC * PLANE) + (size_t)h * NW;  // float idx
    const bool rowInBox = (kp != 0) & (h >= bx) & (h < bx + bs);

    if (!rowInBox) {
        // Fast path: 3 async global->LDS loads in flight, one wait, 3 stores.
        if (active) {
#pragma unroll
            for (int ch = 0; ch < NC; ++ch) {
                const v4f* s = reinterpret_cast<const v4f*>(
                                   images + base + (size_t)ch * PLANE) + lane;
                unsigned ldsAddr = (unsigned)(uintptr_t)&stage[ch][lane];
                unsigned long long ga = (unsigned long long)(uintptr_t)s;
                asm volatile("global_load_async_to_lds_b128 %0, %1, off"
                             :: "v"(ldsAddr), "v"(ga) : "memory");
            }
            asm volatile("s_wait_asynccnt 0x0" ::: "memory");
#pragma unroll
            for (int ch = 0; ch < NC; ++ch) {
                v4f* d = reinterpret_cast<v4f*>(
                             out + base + (size_t)ch * PLANE) + lane;
                unsigned ldsAddr = (unsigned)(uintptr_t)&stage[ch][lane];
                unsigned long long gd = (unsigned long long)(uintptr_t)d;
                asm volatile("global_store_async_from_lds_b128 %0, %1, off"
                             :: "v"(gd), "v"(ldsAddr) : "memory");
            }
            // S_ENDPGM performs an implicit wait-idle; stage[] is not reused.
        }
        return;
    }

    // Blend path: row intersects the box of a kept sample.
    if (active) {
        const size_t sbase = (size_t)pb * (NC * PLANE) + (size_t)h * NW;
        const int w  = lane * 4;
        const int y0 = by, y1 = by + bs;
        const bool m0 = (w + 0 >= y0) & (w + 0 < y1);
        const bool m1 = (w + 1 >= y0) & (w + 1 < y1);
        const bool m2 = (w + 2 >= y0) & (w + 2 < y1);
        const bool m3 = (w + 3 >= y0) & (w + 3 < y1);
#pragma unroll
        for (int ch = 0; ch < NC; ++ch) {
            const v4f a = *(reinterpret_cast<const v4f*>(
                                images + base + (size_t)ch * PLANE) + lane);
            const v4f s = *(reinterpret_cast<const v4f*>(
                                images + sbase + (size_t)ch * PLANE) + lane);
            v4f r;
            r[0] = m0 ? s[0] : a[0];
            r[1] = m1 ? s[1] : a[1];
            r[2] = m2 ? s[2] : a[2];
            r[3] = m3 ? s[3] : a[3];
            v4f* d = reinterpret_cast<v4f*>(out + base + (size_t)ch * PLANE) + lane;
            __builtin_nontemporal_store(r, d);
        }
    }
}

// ---------------------------------------------------------------------------
// Labels: out[b,k] = keep[b] ? lam*labels[b,k] + (1-lam)*labels[perm[b],k]
//                            : labels[b,k],  lam = 1 - box^2/(H*W)
// ---------------------------------------------------------------------------
__global__ __launch_bounds__(256) void cutmix_labels_kernel(
    const float* __restrict__ labels,
    const int*   __restrict__ perm,
    const int*   __restrict__ box_size,
    const unsigned char* __restrict__ keep,
    float* __restrict__ out)
{
    const int i = blockIdx.x * 256 + threadIdx.x;
    if (i >= LBL_FLOATS) return;
    const int b = i / NK;
    const int k = i - b * NK;

    const float l = labels[i];
    float res = l;
    if (keep[b]) {
        const float bs  = (float)box_size[b];
        const float lam = 1.0f - bs * bs * (1.0f / (float)(NH * NW));
        const float ls  = labels[(size_t)perm[b] * NK + k];
        res = lam * l + (1.0f - lam) * ls;
    }
    __builtin_nontemporal_store(res, out + i);
}

// ---------------------------------------------------------------------------
// Launch. Inputs (setup_inputs order):
//   0: images   f32 [128,3,224,224]
//   1: labels   f32 [128,1000]
//   2: perm     i32 [128]
//   3: box_size i32 [128]
//   4: x        i32 [128]
//   5: y        i32 [128]
//   6: keep     bool(1B) [128]
// Output: out_imgs (19,267,584 f32) ++ out_labels (128,000 f32)
// ---------------------------------------------------------------------------
extern "C" void kernel_launch(void* const* d_in, const int* in_sizes, int n_in,
                              void* d_out, int out_size, void* d_ws, size_t ws_size,
                              hipStream_t stream) {
    (void)in_sizes; (void)n_in; (void)out_size; (void)d_ws; (void)ws_size;

    const float* images = (const float*)d_in[0];
    const float* labels = (const float*)d_in[1];
    const int*   perm   = (const int*)d_in[2];
    const int*   boxs   = (const int*)d_in[3];
    const int*   xarr   = (const int*)d_in[4];
    const int*   yarr   = (const int*)d_in[5];
    const unsigned char* keep = (const unsigned char*)d_in[6];

    float* out_imgs   = (float*)d_out;
    float* out_labels = (float*)d_out + IMG_FLOATS;

    const int imgBlocks = NB * NH;                // 28672 (b,h) rows
    cutmix_images_kernel<<<imgBlocks, 64, 0, stream>>>(
        images, perm, boxs, xarr, yarr, keep, out_imgs);

    const int lblBlocks = (LBL_FLOATS + 255) / 256;
    cutmix_labels_kernel<<<lblBlocks, 256, 0, stream>>>(
        labels, perm, boxs, keep, out_labels);
}